// HGNN_5480378269907
// MI455X (gfx1250) — compile-verified
//
#include <hip/hip_runtime.h>
#include <hip/hip_bf16.h>

typedef __attribute__((ext_vector_type(2))) float v2f;
typedef __attribute__((ext_vector_type(8))) float v8f;

#define HID 128

// ---------------------------------------------------------------------------
// float <-> ordered-uint key (monotonic map so atomicMax(uint) == float max)
// ---------------------------------------------------------------------------
__device__ __forceinline__ unsigned f2key(float f) {
    unsigned u = __float_as_uint(f);
    return (u & 0x80000000u) ? ~u : (u | 0x80000000u);
}
__device__ __forceinline__ float key2f(unsigned k) {
    unsigned u = (k & 0x80000000u) ? (k & 0x7FFFFFFFu) : ~k;
    return __uint_as_float(u);
}
#define NEGINF_KEY 0x007FFFFFu   // f2key(-inf)

// ---------------------------------------------------------------------------
// WMMA GEMM: H[N,128] = X[N,128] @ W[128,128], f32 exact via V_WMMA_F32_16X16X4_F32
// MODE 0: store H.  MODE 1: colsum[c] += sum_rows relu(H[r][c] + bias[c])
// Block = 256 threads (8 waves); wave owns a 16-row strip, full 128 columns.
//
// W staged in LDS in k-pair-major layout:  (k,n) -> Wl[(k/2)*256 + n*2 + (k&1)]
// so each B fragment {W[k][c], W[k+1][c]} is one aligned ds_load_b64 landing in
// an adjacent VGPR pair (no repack movs before the WMMA).
// ---------------------------------------------------------------------------
template <int MODE>
__global__ __launch_bounds__(256) void hgnn_gemm128(
    const float* __restrict__ X, const float* __restrict__ W,
    float* __restrict__ H, const float* __restrict__ bias,
    float* __restrict__ colsum, int N)
{
    __shared__ float Wl[HID * HID];
    const int tid = threadIdx.x;
    {
        #pragma unroll 4
        for (int i = tid; i < HID * HID / 4; i += 256) {
            const int k  = i >> 5;          // W row (k index)
            const int n4 = (i & 31) * 4;    // starting column
            const float4 w = ((const float4*)W)[i];
            float* p = &Wl[(k >> 1) * 256 + n4 * 2 + (k & 1)];
            p[0] = w.x; p[2] = w.y; p[4] = w.z; p[6] = w.w;
        }
    }
    __syncthreads();

    const int wave = tid >> 5;
    const int lane = tid & 31;
    const int half = lane >> 4;        // 0: lanes 0-15, 1: lanes 16-31
    const int l16  = lane & 15;
    const int rowBase = blockIdx.x * 128 + wave * 16;
    if (rowBase >= N) return;          // wave-uniform: EXEC stays all-ones

    const float* xrow = X + (size_t)(rowBase + l16) * HID;

    v8f acc[8];
    #pragma unroll
    for (int t = 0; t < 8; ++t) acc[t] = (v8f)0.f;

    // A 16x4 f32 layout: lanes 0-15 row M=l16 K={k0,k0+1}; lanes 16-31 K={k0+2,k0+3}
    // B 4x16 f32 layout (mirror): lanes hold column N=l16, K split by half.
    #pragma unroll 4
    for (int k0 = 0; k0 < HID; k0 += 4) {
        const int ka = k0 + 2 * half;                 // even
        const float2 af = *(const float2*)(xrow + ka);
        v2f A; A.x = af.x; A.y = af.y;
        const float* wbase = &Wl[(ka >> 1) * 256 + l16 * 2];
        #pragma unroll
        for (int t = 0; t < 8; ++t) {
            const v2f B = *(const v2f*)(wbase + t * 32);
            acc[t] = __builtin_amdgcn_wmma_f32_16x16x4_f32(
                false, A, false, B, (short)0, acc[t], false, false);
        }
    }

    // C/D layout: VGPR r -> row (r + 8*half), column t*16 + l16
    if constexpr (MODE == 0) {
        #pragma unroll
        for (int t = 0; t < 8; ++t)
            #pragma unroll
            for (int r = 0; r < 8; ++r)
                H[(size_t)(rowBase + r + 8 * half) * HID + t * 16 + l16] = acc[t][r];
    } else {
        #pragma unroll
        for (int t = 0; t < 8; ++t) {
            const int c = t * 16 + l16;
            const float bc = bias[c];
            float p = 0.f;
            #pragma unroll
            for (int r = 0; r < 8; ++r) p += fmaxf(acc[t][r] + bc, 0.f);
            p += __shfl_xor(p, 16, 32);   // lanes l and l+16 hold same column
            if (half == 0) atomicAdd(&colsum[c], p);
        }
    }
}

// wvec[k] = sum_n W[k][n] * a[n]   (collapses dst projection to a row-dot)
__global__ void hgnn_wvec(const float* __restrict__ W, const float* __restrict__ a,
                          float* __restrict__ wvec)
{
    const int k = threadIdx.x;
    float s = 0.f;
    #pragma unroll 4
    for (int n = 0; n < HID; ++n) s += W[k * HID + n] * a[n];
    wvec[k] = s;
}

// al[i] = dot(X[i,:], v)   one wave per row
__global__ __launch_bounds__(256) void hgnn_rowdot(
    const float* __restrict__ X, const float* __restrict__ v,
    float* __restrict__ al, int N)
{
    const int row  = blockIdx.x * 8 + (threadIdx.x >> 5);
    const int lane = threadIdx.x & 31;
    if (row >= N) return;
    const float4 xv = ((const float4*)(X + (size_t)row * HID))[lane];
    const float4 vv = ((const float4*)v)[lane];
    float s = xv.x * vv.x + xv.y * vv.y + xv.z * vv.z + xv.w * vv.w;
    #pragma unroll
    for (int off = 16; off > 0; off >>= 1) s += __shfl_xor(s, off, 32);
    if (lane == 0) al[row] = s;
}

__global__ void hgnn_fill_f32(float* __restrict__ p, float v, int n) {
    const int i = blockIdx.x * 256 + threadIdx.x;
    if (i < n) p[i] = v;
}
__global__ void hgnn_fill_u32(unsigned* __restrict__ p, unsigned v, int n) {
    const int i = blockIdx.x * 256 + threadIdx.x;
    if (i < n) p[i] = v;
}
// out[i][c] = b[c]  (bias broadcast prior to edge scatter)
__global__ void hgnn_bias_init(float* __restrict__ out, const float* __restrict__ b, int n) {
    const int i = blockIdx.x * 256 + threadIdx.x;
    if (i < n) out[i] = b[i & (HID - 1)];
}

__device__ __forceinline__ float edge_logit(const float* al_s, const float* al_d,
                                            int s, int d) {
    float v = al_s[s] + al_d[d];
    return v > 0.f ? v : 0.2f * v;      // leaky_relu(., 0.2)
}

__global__ void hgnn_edge_max(const float* __restrict__ al_s, const float* __restrict__ al_d,
                              const int* __restrict__ src, const int* __restrict__ dst,
                              unsigned* __restrict__ mkey, int E)
{
    const int e = blockIdx.x * 256 + threadIdx.x;
    if (e >= E) return;
    const int s = src[e], d = dst[e];
    atomicMax(&mkey[d], f2key(edge_logit(al_s, al_d, s, d)));
}

__global__ void hgnn_edge_sum(const float* __restrict__ al_s, const float* __restrict__ al_d,
                              const int* __restrict__ src, const int* __restrict__ dst,
                              const unsigned* __restrict__ mkey, float* __restrict__ den, int E)
{
    const int e = blockIdx.x * 256 + threadIdx.x;
    if (e >= E) return;
    const int s = src[e], d = dst[e];
    const float v = edge_logit(al_s, al_d, s, d);
    atomicAdd(&den[d], __expf(v - key2f(mkey[d])));
}

// One wave per edge: out[dst] += alpha * hs[src]  (4 floats per lane)
__global__ __launch_bounds__(256) void hgnn_edge_scatter(
    const float* __restrict__ al_s, const float* __restrict__ al_d,
    const int* __restrict__ src, const int* __restrict__ dst,
    const unsigned* __restrict__ mkey, const float* __restrict__ den,
    const float* __restrict__ hs, float* __restrict__ out, int E)
{
    const int e    = blockIdx.x * 8 + (threadIdx.x >> 5);
    const int lane = threadIdx.x & 31;
    if (e >= E) return;
    const int s = src[e], d = dst[e];
    float alpha = 0.f;
    if (lane == 0) {
        const float v  = edge_logit(al_s, al_d, s, d);
        const float ex = __expf(v - key2f(mkey[d]));
        alpha = ex / (den[d] + 1e-16f);
    }
    alpha = __shfl(alpha, 0, 32);
    const float4 h = ((const float4*)(hs + (size_t)s * HID))[lane];
    float* o = out + (size_t)d * HID + lane * 4;
    atomicAdd(o + 0, alpha * h.x);
    atomicAdd(o + 1, alpha * h.y);
    atomicAdd(o + 2, alpha * h.z);
    atomicAdd(o + 3, alpha * h.w);
}

__global__ void hgnn_relu(float* __restrict__ x, int n) {
    const int i = blockIdx.x * 256 + threadIdx.x;
    if (i < n) x[i] = fmaxf(x[i], 0.f);
}
// a = relu(0.5*(a+b))  (aggr='mean' of the two rd convolutions)
__global__ void hgnn_combine_rd(float* __restrict__ a, const float* __restrict__ b, int n) {
    const int i = blockIdx.x * 256 + threadIdx.x;
    if (i < n) a[i] = fmaxf(0.5f * (a[i] + b[i]), 0.f);
}

// Final head: z = concat(colsum/count); softmax(z @ W_fc + b_fc)
__global__ void hgnn_head(const float* __restrict__ colsum,
                          float cA, float cS, float cD, float cT,
                          const float* __restrict__ W_fc, const float* __restrict__ b_fc,
                          float* __restrict__ out)
{
    __shared__ float z[512];
    __shared__ float lg[16];
    const int t = threadIdx.x;
    const float cnt[4] = {cA, cS, cD, cT};
    if (t < 512) z[t] = colsum[t] / cnt[t >> 7];
    __syncthreads();
    if (t < 16) {
        float s = b_fc[t];
        for (int j = 0; j < 512; ++j) s += z[j] * W_fc[j * 16 + t];
        lg[t] = s;
    }
    __syncthreads();
    if (t == 0) {
        float mx = lg[0];
        for (int i = 1; i < 16; ++i) mx = fmaxf(mx, lg[i]);
        float se = 0.f, e[16];
        for (int i = 0; i < 16; ++i) { e[i] = __expf(lg[i] - mx); se += e[i]; }
        for (int i = 0; i < 16; ++i) out[i] = e[i] / se;
    }
}

// ---------------------------------------------------------------------------
extern "C" void kernel_launch(void* const* d_in, const int* in_sizes, int n_in,
                              void* d_out, int out_size, void* d_ws, size_t ws_size,
                              hipStream_t stream)
{
    (void)in_sizes; (void)n_in; (void)out_size; (void)ws_size;
    const int NA = 200000, NS = 50000, ND = 100000, NT = 100000;
    const int ECNT[5] = {600000, 400000, 500000, 500000, 500000};

    const float* x_act  = (const float*)d_in[0];
    const float* x_rs   = (const float*)d_in[1];
    const float* x_rd   = (const float*)d_in[2];
    const float* x_attr = (const float*)d_in[3];
    const int* EI[5] = {(const int*)d_in[4], (const int*)d_in[5], (const int*)d_in[6],
                        (const int*)d_in[7], (const int*)d_in[8]};
    const float* W_src   = (const float*)d_in[9];
    const float* W_dst   = (const float*)d_in[10];
    const float* att_src = (const float*)d_in[11];
    const float* att_dst = (const float*)d_in[12];
    const float* b_gat   = (const float*)d_in[13];
    const float* W_ln    = (const float*)d_in[14];
    const float* b_ln    = (const float*)d_in[15];
    const float* W_fc    = (const float*)d_in[16];
    const float* b_fc    = (const float*)d_in[17];

    float* ws = (float*)d_ws;
    size_t off = 0;
    auto take = [&](size_t n) { float* p = ws + off; off += n; return p; };
    float* xB[4] = {take((size_t)NA * HID), take((size_t)NS * HID),
                    take((size_t)ND * HID), take((size_t)NT * HID)};
    float* xC[4] = {take((size_t)NA * HID), take((size_t)NS * HID),
                    take((size_t)ND * HID), take((size_t)NT * HID)};
    float* hs     = take((size_t)NA * HID);
    float* rdB    = take((size_t)ND * HID);
    float* al_s   = take(NA);
    float* al_d   = take(NA);
    unsigned* mkey = (unsigned*)take(NA);
    float* den    = take(NA);
    float* wvec   = take(HID);
    float* colsum = take(512);

    for (int l = 0; l < 2; ++l) {
        const float* cur[4];
        if (l == 0) { cur[0] = x_act; cur[1] = x_rs; cur[2] = x_rd; cur[3] = x_attr; }
        else        { cur[0] = xB[0]; cur[1] = xB[1]; cur[2] = xB[2]; cur[3] = xB[3]; }
        float* const* nxt = (l == 0) ? xB : xC;

        struct R { const float* xs; int ns; const float* xd; int nd; int ei; float* out; };
        const R rels[5] = {
            {cur[0], NA, cur[0], NA, 0, nxt[0]},
            {cur[0], NA, cur[1], NS, 1, nxt[1]},
            {cur[2], ND, cur[2], ND, 2, nxt[2]},
            {cur[0], NA, cur[2], ND, 3, rdB},
            {cur[0], NA, cur[3], NT, 4, nxt[3]},
        };

        for (int r = 0; r < 5; ++r) {
            const R& q = rels[r];
            const float* Ws  = W_src   + (size_t)(l * 5 + r) * HID * HID;
            const float* Wd  = W_dst   + (size_t)(l * 5 + r) * HID * HID;
            const float* as_ = att_src + (size_t)(l * 5 + r) * HID;
            const float* ad_ = att_dst + (size_t)(l * 5 + r) * HID;
            const float* bg  = b_gat   + (size_t)(l * 5 + r) * HID;
            const int* src = EI[q.ei];
            const int* dst = src + ECNT[q.ei];
            const int E = ECNT[q.ei];

            hgnn_gemm128<0><<<(q.ns + 127) / 128, 256, 0, stream>>>(
                q.xs, Ws, hs, nullptr, nullptr, q.ns);
            hgnn_rowdot<<<(q.ns + 7) / 8, 256, 0, stream>>>(hs, as_, al_s, q.ns);
            hgnn_wvec<<<1, 128, 0, stream>>>(Wd, ad_, wvec);
            hgnn_rowdot<<<(q.nd + 7) / 8, 256, 0, stream>>>(q.xd, wvec, al_d, q.nd);
            hgnn_fill_u32<<<(q.nd + 255) / 256, 256, 0, stream>>>(mkey, NEGINF_KEY, q.nd);
            hgnn_fill_f32<<<(q.nd + 255) / 256, 256, 0, stream>>>(den, 0.f, q.nd);
            hgnn_edge_max<<<(E + 255) / 256, 256, 0, stream>>>(al_s, al_d, src, dst, mkey, E);
            hgnn_edge_sum<<<(E + 255) / 256, 256, 0, stream>>>(al_s, al_d, src, dst, mkey, den, E);
            hgnn_bias_init<<<(q.nd * HID + 255) / 256, 256, 0, stream>>>(q.out, bg, q.nd * HID);
            hgnn_edge_scatter<<<(E + 7) / 8, 256, 0, stream>>>(
                al_s, al_d, src, dst, mkey, den, hs, q.out, E);
        }

        hgnn_relu<<<(NA * HID + 255) / 256, 256, 0, stream>>>(nxt[0], NA * HID);
        hgnn_relu<<<(NS * HID + 255) / 256, 256, 0, stream>>>(nxt[1], NS * HID);
        hgnn_combine_rd<<<(ND * HID + 255) / 256, 256, 0, stream>>>(nxt[2], rdB, ND * HID);
        hgnn_relu<<<(NT * HID + 255) / 256, 256, 0, stream>>>(nxt[3], NT * HID);
    }

    // Pooled head: colsum[s][c] = sum_rows relu(x @ W_ln + b_ln)
    hgnn_fill_f32<<<2, 256, 0, stream>>>(colsum, 0.f, 512);
    const float* fx[4] = {xC[0], xC[1], xC[2], xC[3]};
    const int    nN[4] = {NA, NS, ND, NT};
    for (int s2 = 0; s2 < 4; ++s2)
        hgnn_gemm128<1><<<(nN[s2] + 127) / 128, 256, 0, stream>>>(
            fx[s2], W_ln, nullptr, b_ln, colsum + s2 * HID, nN[s2]);
    hgnn_head<<<1, 512, 0, stream>>>(colsum, (float)NA, (float)NS, (float)ND, (float)NT,
                                     W_fc, b_fc, (float*)d_out);
}